// MultiresolutionalHashEncoding_6021544149816
// MI455X (gfx1250) — compile-verified
//
#include <hip/hip_runtime.h>
#include <stdint.h>

#define NLEVEL 16
#define MODT   65536
#define HH     256
#define WW     256
#define NB     16

// Static level geometry (levels 0..12 go through workspace; 13..15 are fused identity)
// P[l] = 256 // (256 // grid[l]) per reduce_window VALID semantics.
__device__ __constant__ int c_P[13]    = {4, 5, 6, 9, 12, 16, 21, 28, 36, 51, 64, 85, 128};
__device__ __constant__ int c_base[13] = {0, 768, 1968, 3696, 7584, 14496, 26784, 47952,
                                          85584, 147792, 272640, 469248, 816048};
// total workspace: 48 * 33385 = 1,602,480 floats = 6.4 MB

__device__ __forceinline__ uint32_t hash3(uint32_t u0, uint32_t u1, uint32_t u2) {
    // int64 products XORed then %65536  ==  u32 wrap-around products XORed & 0xFFFF
    return (u0 ^ (u1 * 2654435761u) ^ (u2 * 805459861u)) & 0xFFFFu;
}

// ---------------------------------------------------------------------------
// Coarse levels (K >= 16): one block per pooled cell. Stage the KxK window
// into LDS with CDNA5 async global->LDS copies (ASYNCcnt path), fence with
// s_wait_asynccnt, then tree-reduce the max in LDS.
// ---------------------------------------------------------------------------
__global__ void pool_hash_block(const float* __restrict__ img,
                                const float* __restrict__ table,
                                float* __restrict__ F,
                                int lvl, float gridf, int K, int P, unsigned PP) {
    __shared__ float tile[64 * 64];   // up to 16 KB window
    __shared__ float red[256];
    __shared__ float chMax[3];

    const unsigned cell = blockIdx.x;          // over NB*P*P
    const unsigned b    = cell / PP;
    const unsigned pyx  = cell - b * PP;
    const unsigned py   = pyx / (unsigned)P;
    const unsigned px   = pyx - py * (unsigned)P;
    const int y0   = (int)py * K;
    const int x0   = (int)px * K;
    const int tid  = threadIdx.x;
    const int n    = K * K;

    for (int c = 0; c < 3; ++c) {
        const float* src = img + ((size_t)(b * 3 + c) * HH + y0) * WW + x0;
        // async-stage window into LDS (per-lane copy, no VGPR round trip)
        for (int i = tid; i < n; i += 256) {
            int r  = i / K;
            int cc = i - r * K;
            uint32_t lds_addr = (uint32_t)(uintptr_t)&tile[i];      // low 32 bits = LDS offset
            uint32_t voff     = (uint32_t)((r * WW + cc) * 4);
            asm volatile("global_load_async_to_lds_b32 %0, %1, %2"
                         :: "v"(lds_addr), "v"(voff), "s"(src)
                         : "memory");
        }
        asm volatile("s_wait_asynccnt 0x0" ::: "memory");   // this wave's copies done
        __syncthreads();                                    // all waves' copies done

        float m = -3.402823466e38f;
        for (int i = tid; i < n; i += 256) m = fmaxf(m, tile[i]);
        red[tid] = m;
        __syncthreads();
        #pragma unroll
        for (int s = 128; s > 0; s >>= 1) {
            if (tid < s) red[tid] = fmaxf(red[tid], red[tid + s]);
            __syncthreads();
        }
        if (tid == 0) chMax[c] = red[0];
        __syncthreads();
    }

    if (tid == 0) {
        uint32_t u0 = (uint32_t)(int)(chMax[0] * gridf);
        uint32_t u1 = (uint32_t)(int)(chMax[1] * gridf);
        uint32_t u2 = (uint32_t)(int)(chMax[2] * gridf);
        uint32_t g  = hash3(u0, u1, u2);
        const float* t = table + ((size_t)lvl * MODT + g) * 3;
        unsigned base = (b * 3) * PP + py * (unsigned)P + px;   // layout (b,3,P,P)
        F[base]          = t[0];
        F[base + PP]     = t[1];
        F[base + 2 * PP] = t[2];
    }
}

// ---------------------------------------------------------------------------
// Fine levels (K < 16): one thread per pooled cell (L2-resident reads).
// ---------------------------------------------------------------------------
__global__ void pool_hash_thread(const float* __restrict__ img,
                                 const float* __restrict__ table,
                                 float* __restrict__ F,
                                 int lvl, float gridf, int K, int P, unsigned PP) {
    unsigned idx = blockIdx.x * blockDim.x + threadIdx.x;
    unsigned total = NB * PP;
    if (idx >= total) return;
    unsigned b   = idx / PP;
    unsigned pyx = idx - b * PP;
    unsigned py  = pyx / (unsigned)P;
    unsigned px  = pyx - py * (unsigned)P;
    int y0  = (int)py * K;
    int x0  = (int)px * K;

    uint32_t u[3];
    for (int c = 0; c < 3; ++c) {
        const float* src = img + ((size_t)(b * 3 + c) * HH + y0) * WW + x0;
        float m = -3.402823466e38f;
        for (int r = 0; r < K; ++r)
            for (int cc = 0; cc < K; ++cc)
                m = fmaxf(m, src[r * WW + cc]);
        u[c] = (uint32_t)(int)(m * gridf);
    }
    uint32_t g = hash3(u[0], u[1], u[2]);
    const float* t = table + ((size_t)lvl * MODT + g) * 3;
    unsigned base = (b * 3) * PP + py * (unsigned)P + px;
    F[base]          = t[0];
    F[base + PP]     = t[1];
    F[base + 2 * PP] = t[2];
}

// ---------------------------------------------------------------------------
// Fused bilinear upsample for ALL 13 workspace levels. blockIdx.y = level.
// Half-pixel centers, edge clamp. Output stored non-temporally (write-once,
// never re-read -> keep L2 for image + hash table).
// ---------------------------------------------------------------------------
__global__ void upsample_all(const float* __restrict__ ws,
                             float* __restrict__ out) {
    unsigned idx = blockIdx.x * blockDim.x + threadIdx.x;   // over NB*3*H*W
    if (idx >= NB * 3 * HH * WW) return;
    int lvl = blockIdx.y;
    int P   = c_P[lvl];
    const float* F = ws + c_base[lvl];

    unsigned b = idx / (3u * HH * WW);
    unsigned q = idx - b * (3u * HH * WW);
    unsigned c = q >> 16;
    unsigned y = (q >> 8) & 255u;
    unsigned x = q & 255u;

    float scale = (float)P * (1.0f / 256.0f);
    float fy = ((float)y + 0.5f) * scale - 0.5f;
    float fx = ((float)x + 0.5f) * scale - 0.5f;
    int iy0 = (int)floorf(fy);
    int ix0 = (int)floorf(fx);
    float wy = fy - (float)iy0;
    float wx = fx - (float)ix0;
    int y0c = min(max(iy0, 0), P - 1);
    int y1c = min(max(iy0 + 1, 0), P - 1);
    int x0c = min(max(ix0, 0), P - 1);
    int x1c = min(max(ix0 + 1, 0), P - 1);

    const float* f = F + (size_t)(b * 3 + c) * P * P;
    float v00 = f[y0c * P + x0c];
    float v01 = f[y0c * P + x1c];
    float v10 = f[y1c * P + x0c];
    float v11 = f[y1c * P + x1c];
    float v = (1.0f - wy) * ((1.0f - wx) * v00 + wx * v01)
            +         wy  * ((1.0f - wx) * v10 + wx * v11);

    float* dst = &out[(((size_t)b * 48 + lvl * 3 + c) * HH + y) * WW + x];
    __builtin_nontemporal_store(v, dst);
}

// ---------------------------------------------------------------------------
// Levels 13..15: K==1 so pooling is identity and P==256 so resize is identity.
// Fully fused: one thread per pixel, reads image once for all three levels.
// ---------------------------------------------------------------------------
__global__ void fused_fine_levels(const float* __restrict__ img,
                                  const float* __restrict__ table,
                                  float* __restrict__ out) {
    unsigned idx = blockIdx.x * blockDim.x + threadIdx.x;   // over NB*H*W
    if (idx >= NB * HH * WW) return;
    unsigned b = idx / (HH * WW);
    unsigned p = idx - b * (HH * WW);

    size_t ib = (size_t)b * 3 * HH * WW + p;
    float v0 = img[ib];
    float v1 = img[ib + HH * WW];
    float v2 = img[ib + 2 * HH * WW];

    const int   lvls[3] = {13, 14, 15};
    const float gr[3]   = {147.0f, 194.0f, 256.0f};
    #pragma unroll
    for (int i = 0; i < 3; ++i) {
        uint32_t u0 = (uint32_t)(int)(v0 * gr[i]);
        uint32_t u1 = (uint32_t)(int)(v1 * gr[i]);
        uint32_t u2 = (uint32_t)(int)(v2 * gr[i]);
        uint32_t g  = hash3(u0, u1, u2);
        const float* t = table + ((size_t)lvls[i] * MODT + g) * 3;
        size_t ob = ((size_t)b * 48 + lvls[i] * 3) * (size_t)(HH * WW) + p;
        __builtin_nontemporal_store(t[0], &out[ob]);
        __builtin_nontemporal_store(t[1], &out[ob + HH * WW]);
        __builtin_nontemporal_store(t[2], &out[ob + 2 * HH * WW]);
    }
}

extern "C" void kernel_launch(void* const* d_in, const int* in_sizes, int n_in,
                              void* d_out, int out_size, void* d_ws, size_t ws_size,
                              hipStream_t stream) {
    (void)in_sizes; (void)n_in; (void)out_size; (void)ws_size;
    const float* img   = (const float*)d_in[0];
    const float* table = (const float*)d_in[1];
    float*       out   = (float*)d_out;
    float*       ws    = (float*)d_ws;

    static const int grids[NLEVEL] = {4, 5, 6, 9, 12, 16, 21, 27, 36, 48,
                                      64, 84, 111, 147, 194, 256};
    static const int bases[13]     = {0, 768, 1968, 3696, 7584, 14496, 26784, 47952,
                                      85584, 147792, 272640, 469248, 816048};

    // Phase 1: pooled-hash-gather into workspace, levels 0..12
    for (int l = 0; l < 13; ++l) {
        int K = 256 / grids[l];
        int P = 256 / K;
        unsigned PP = (unsigned)(P * P);
        float* F = ws + bases[l];

        if (K >= 16) {
            pool_hash_block<<<NB * P * P, 256, 0, stream>>>(
                img, table, F, l, (float)grids[l], K, P, PP);
        } else {
            int total = NB * P * P;
            pool_hash_thread<<<(total + 255) / 256, 256, 0, stream>>>(
                img, table, F, l, (float)grids[l], K, P, PP);
        }
    }

    // Phase 2: one fused bilinear-upsample kernel over all 13 levels
    {
        int totalUp = NB * 3 * HH * WW;
        dim3 grid((totalUp + 255) / 256, 13, 1);
        upsample_all<<<grid, 256, 0, stream>>>(ws, out);
    }

    // Phase 3: identity levels 13..15, fully fused
    {
        int totalF = NB * HH * WW;
        fused_fine_levels<<<(totalF + 255) / 256, 256, 0, stream>>>(img, table, out);
    }
}